// PWLToneMapping_86732569575706
// MI455X (gfx1250) — compile-verified
//
#include <hip/hip_runtime.h>
#include <stdint.h>

// PWL tone map, 4096x8192 fp32. Memory-bound: 268 MB @ 23.3 TB/s ~= 11.5 us floor.
// Prep kernel (1 block): sort breakpoints, build interleaved slope/intercept and a
// 1024-bucket uint8 "count <= bucket-left-edge" table. Apply kernel: stage the
// 2.2 KB tables into LDS via gfx1250 async-to-LDS DMA, stream image with float4.
//
// ws layout (dwords): [0..99] sorted_x, [100..299] {slope,intercept} pairs (float2),
// [300..555] 1024 x uint8 cnt0.

#define K_BP    100
#define NBUCKET 1024
#define TAB_DW  (3 * K_BP + NBUCKET / 4)   // 556 dwords = 2224 bytes

#if defined(__has_builtin)
#if __has_builtin(__builtin_amdgcn_global_load_async_to_lds_b32) && \
    __has_builtin(__builtin_amdgcn_s_wait_asynccnt)
#define USE_ASYNC_TABLE 1
#endif
#endif

// ---------------------------------------------------------------------------
// Prep: sort breakpoints, build tables into d_ws.
// ---------------------------------------------------------------------------
__global__ __launch_bounds__(256) void pwl_prep_kernel(
    const float* __restrict__ xp, const float* __restrict__ sl,
    const float* __restrict__ bias, float* __restrict__ ws)
{
    __shared__ float sx[K_BP];
    __shared__ float betas[K_BP];
    const int t = threadIdx.x;

    // Rank sort (stable): rank = #(u < v) + #(u == v with smaller index).
    if (t < K_BP) {
        float v = xp[t];
        int r = 0;
        for (int j = 0; j < K_BP; ++j) {
            float u = xp[j];
            r += ((u < v) || ((u == v) && (j < t))) ? 1 : 0;
        }
        sx[r] = v;
    }
    __syncthreads();

    // betas[i] = bias + sum_{j<i} (sx[j+1]-sx[j]) * slope[j]  (sequential, K=100)
    if (t == 0) {
        float acc = bias[0];
        betas[0] = acc;
        for (int i = 0; i < K_BP - 1; ++i) {
            acc = fmaf(sx[i + 1] - sx[i], sl[i], acc);
            betas[i + 1] = acc;
        }
    }
    __syncthreads();

    if (t < K_BP) {
        float m = sl[t];            // slopes indexed by sorted position (ref semantics)
        float s = sx[t];
        ws[t] = s;
        ws[K_BP + 2 * t]     = m;                     // interleaved {slope, intercept}
        ws[K_BP + 2 * t + 1] = fmaf(-s, m, betas[t]); // intercept: y = m*x + c
    }

    // Bucket table: cnt0[b] = #(sx <= b/1024). b/1024 is exact in fp32.
    uint8_t* cnt0 = (uint8_t*)(ws + 3 * K_BP);
    for (int b = t; b < NBUCKET; b += 256) {
        float L = (float)b * (1.0f / (float)NBUCKET);
        int c = 0;
        for (int j = 0; j < K_BP; ++j) c += (sx[j] <= L) ? 1 : 0;
        cnt0[b] = (uint8_t)c;
    }
}

// ---------------------------------------------------------------------------
// Per-element PWL evaluation from LDS tables.
// ---------------------------------------------------------------------------
__device__ __forceinline__ float pwl_one(
    float x, const float* __restrict__ sx, const float2* __restrict__ mc,
    const uint8_t* __restrict__ cnt0)
{
    // x*1024 is a power-of-two multiply -> exact; floor gives bucket.
    int b = (int)(x * (float)NBUCKET);
    b = (b < 0) ? 0 : ((b > NBUCKET - 1) ? (NBUCKET - 1) : b);
    int c = cnt0[b];                       // count(sx <= bucket_left) <= count(sx <= x)
    #pragma unroll 1
    while (c < K_BP && sx[c] <= x) ++c;    // finish the count (avg ~0.1 iterations)
    int idx = c;                           // idx = clip(count, 1, K) - 1
    if (idx < 1) idx = 1;
    if (idx > K_BP) idx = K_BP;
    --idx;
    float2 p = mc[idx];                    // single ds_load_b64: {slope, intercept}
    float y = fmaf(x, p.x, p.y);
    return fminf(fmaxf(y, 0.0f), 1.0f);
}

// ---------------------------------------------------------------------------
// Apply: stage tables into LDS, stream image with float4.
// ---------------------------------------------------------------------------
__global__ __launch_bounds__(256) void pwl_apply_kernel(
    const float* __restrict__ in, float* __restrict__ out,
    const float* __restrict__ ws, int n)
{
    __shared__ __align__(16) uint32_t s_all[TAB_DW];
    const uint32_t* wsw = (const uint32_t*)ws;
    const int t = threadIdx.x;

#ifdef USE_ASYNC_TABLE
    // gfx1250 async DMA into LDS, tracked by ASYNCcnt (not LOADcnt).
    for (int i = t; i < TAB_DW; i += 256) {
        __builtin_amdgcn_global_load_async_to_lds_b32(
            (__attribute__((address_space(1))) int*)(wsw + i),
            (__attribute__((address_space(3))) int*)(&s_all[i]),
            0, 0);
    }
    __builtin_amdgcn_s_wait_asynccnt(0);
    __syncthreads();
#else
    for (int i = t; i < TAB_DW; i += 256) s_all[i] = wsw[i];
    __syncthreads();
#endif

    const float*   s_sx  = (const float*)s_all;                 // LDS byte 0
    const float2*  s_mc  = (const float2*)(s_all + K_BP);       // LDS byte 400 (8B aligned)
    const uint8_t* s_cnt = (const uint8_t*)(s_all + 3 * K_BP);  // LDS byte 1200

    const float4* in4  = (const float4*)in;
    float4*       out4 = (float4*)out;
    const int n4     = n >> 2;
    const int stride = (int)gridDim.x * 256;

    for (int i = (int)blockIdx.x * 256 + t; i < n4; i += stride) {
        float4 v = in4[i];
        if (i + stride < n4) __builtin_prefetch(&in4[i + stride], 0, 0);
        float4 r;
        r.x = pwl_one(v.x, s_sx, s_mc, s_cnt);
        r.y = pwl_one(v.y, s_sx, s_mc, s_cnt);
        r.z = pwl_one(v.z, s_sx, s_mc, s_cnt);
        r.w = pwl_one(v.w, s_sx, s_mc, s_cnt);
        out4[i] = r;
    }

    // Scalar tail (n % 4), handled by the first few global threads.
    const int base = n4 << 2;
    const int gid  = (int)blockIdx.x * 256 + t;
    if (gid < (n - base)) {
        out[base + gid] = pwl_one(in[base + gid], s_sx, s_mc, s_cnt);
    }
}

// ---------------------------------------------------------------------------
// Host entry. Inputs: x (B*C*H*W), x_positions (100), slopes (100), biases (1).
// ---------------------------------------------------------------------------
extern "C" void kernel_launch(void* const* d_in, const int* in_sizes, int n_in,
                              void* d_out, int out_size, void* d_ws, size_t ws_size,
                              hipStream_t stream)
{
    const float* x    = (const float*)d_in[0];
    const float* xp   = (const float*)d_in[1];
    const float* sl   = (const float*)d_in[2];
    const float* bias = (const float*)d_in[3];
    float* out = (float*)d_out;
    float* ws  = (float*)d_ws;   // needs TAB_DW*4 = 2224 bytes of scratch

    const int n = out_size;

    pwl_prep_kernel<<<1, 256, 0, stream>>>(xp, sl, bias, ws);

    const int n4 = n >> 2;
    const int work_per_block = 256 * 4;              // 4 float4 per thread
    int blocks = (n4 + work_per_block - 1) / work_per_block;
    if (blocks < 1) blocks = 1;
    pwl_apply_kernel<<<blocks, 256, 0, stream>>>(x, out, ws, n);
}